// PrecRec_19284403159419
// MI455X (gfx1250) — compile-verified
//
#include <hip/hip_runtime.h>
#include <hip/hip_bf16.h>
#include <stdint.h>

// Precision/Recall confusion counts on MI455X (gfx1250, wave32).
// Streaming phase is HBM-bound (~402MB @ 23.3TB/s ~= 17us floor), so the
// kernel minimizes per-element ALU with logit-transformed thresholds and
// wave32 ballot+popcount (scalar-pipe) accumulation. The final 4x10 output
// is a constant 4x4 linear map of (tp, p, sum_m, sum_mt) and is computed
// with a single v_wmma_f32_16x16x4_f32.

#define NT 10

__device__ __forceinline__ unsigned ballot32(bool p) {
#if __has_builtin(__builtin_amdgcn_ballot_w32)
  return __builtin_amdgcn_ballot_w32(p);
#else
  return (unsigned)__ballot(p);
#endif
}

__global__ void precrec_count_kernel(const float* __restrict__ pred,
                                     const int* __restrict__ mask,
                                     const int* __restrict__ targ,
                                     unsigned* __restrict__ ws,
                                     long long n) {
  // logit(k/11), k=1..10: sigmoid(x) > k/11  <=>  x > logit(k/11)
  constexpr float LG[NT] = {
      -2.302585093f, -1.504077397f, -0.980829253f, -0.559615788f,
      -0.182321557f,  0.182321557f,  0.559615788f,  0.980829253f,
       1.504077397f,  2.302585093f};

  __shared__ unsigned sh[2 * NT + 2];
  if (threadIdx.x < 2 * NT + 2) sh[threadIdx.x] = 0u;
  __syncthreads();

  // Wave-uniform accumulators (ballot results are uniform across the wave).
  unsigned pc[NT];
  unsigned tpc[NT];
#pragma unroll
  for (int k = 0; k < NT; ++k) { pc[k] = 0u; tpc[k] = 0u; }
  unsigned tm = 0u, tmt = 0u;

  const long long nv = n >> 2;  // float4 / int4 vectors
  const long long stride = (long long)gridDim.x * blockDim.x;
  const float4* __restrict__ p4 = reinterpret_cast<const float4*>(pred);
  const int4*   __restrict__ m4 = reinterpret_cast<const int4*>(mask);
  const int4*   __restrict__ t4 = reinterpret_cast<const int4*>(targ);

  for (long long i = (long long)blockIdx.x * blockDim.x + threadIdx.x;
       i < nv; i += stride) {
    if (i + stride < nv) {
      // gfx1250 global_prefetch_b8 for the next grid-stride chunk
      __builtin_prefetch(&p4[i + stride], 0, 0);
      __builtin_prefetch(&m4[i + stride], 0, 0);
      __builtin_prefetch(&t4[i + stride], 0, 0);
    }
    const float4 x  = p4[i];
    const int4   mm = m4[i];
    const int4   tt = t4[i];
    const float xs[4] = {x.x, x.y, x.z, x.w};
    const int   ms[4] = {mm.x, mm.y, mm.z, mm.w};
    const int   ts[4] = {tt.x, tt.y, tt.z, tt.w};
#pragma unroll
    for (int j = 0; j < 4; ++j) {
      const bool mf  = (ms[j] != 0);
      const bool mtf = mf && (ts[j] != 0);
      // Bias invalid lanes below every threshold: one cndmask each.
      const float xm  = mf  ? xs[j] : -3.0e38f;
      const float xmt = mtf ? xs[j] : -3.0e38f;
      tm  += (unsigned)__builtin_popcount(ballot32(mf));
      tmt += (unsigned)__builtin_popcount(ballot32(mtf));
#pragma unroll
      for (int k = 0; k < NT; ++k) {
        pc[k]  += (unsigned)__builtin_popcount(ballot32(xm  > LG[k]));
        tpc[k] += (unsigned)__builtin_popcount(ballot32(xmt > LG[k]));
      }
    }
  }

  // Tail (n % 4) — one wave of block 0, still via ballot so totals stay
  // wave-uniform. (For this problem n % 4 == 0, kept for generality.)
  const long long base = nv << 2;
  if (blockIdx.x == 0 && threadIdx.x < 32) {
    for (long long e = base; e < n; e += 32) {
      const long long idx = e + threadIdx.x;
      const bool valid = idx < n;
      const float xv = valid ? pred[idx] : 0.0f;
      const int   mv = valid ? mask[idx] : 0;
      const int   tv = valid ? targ[idx] : 0;
      const bool mf  = valid && (mv != 0);
      const bool mtf = mf && (tv != 0);
      const float xm  = mf  ? xv : -3.0e38f;
      const float xmt = mtf ? xv : -3.0e38f;
      tm  += (unsigned)__builtin_popcount(ballot32(mf));
      tmt += (unsigned)__builtin_popcount(ballot32(mtf));
#pragma unroll
      for (int k = 0; k < NT; ++k) {
        pc[k]  += (unsigned)__builtin_popcount(ballot32(xm  > LG[k]));
        tpc[k] += (unsigned)__builtin_popcount(ballot32(xmt > LG[k]));
      }
    }
  }

  // Per-block merge: wave leaders -> LDS, then block -> global atomics.
  if ((threadIdx.x & 31u) == 0u) {
#pragma unroll
    for (int k = 0; k < NT; ++k) {
      atomicAdd(&sh[k], pc[k]);
      atomicAdd(&sh[NT + k], tpc[k]);
    }
    atomicAdd(&sh[2 * NT], tm);
    atomicAdd(&sh[2 * NT + 1], tmt);
  }
  __syncthreads();
  if (threadIdx.x < 2 * NT + 2) atomicAdd(&ws[threadIdx.x], sh[threadIdx.x]);
}

// ---------------------------------------------------------------------------
// Finalize: OUT(4x10) = W(4x4) * X(4x10) with
//   X = [tp; p; sum_m * 1; sum_mt * 1]
//   W rows: tp=[1,0,0,0], fp=[-1,1,0,0], tn=[1,-1,1,-1], fn=[-1,0,0,1]
// computed by one V_WMMA_F32_16X16X4_F32 (W in A[16x4] rows 0..3, X in
// B[4x16] cols 0..9, C = 0). Exact in f32 (counts < 2^25).
// ---------------------------------------------------------------------------
typedef __attribute__((ext_vector_type(2))) float v2f;
typedef __attribute__((ext_vector_type(8))) float v8f;

__global__ void precrec_finalize_kernel(const unsigned* __restrict__ ws,
                                        float* __restrict__ out) {
#if __has_builtin(__builtin_amdgcn_wmma_f32_16x16x4_f32)
  const int lane = threadIdx.x;   // one full wave, EXEC all ones for WMMA
  const int nc   = lane & 15;     // column index (N)
  const int hi   = lane >> 4;     // half-wave selects K pair {0,1} vs {2,3}

  // Gather X column nc with clamped (non-divergent) loads, then zero cols>=10.
  const int ci = nc < NT ? nc : NT - 1;
  float tpv  = (float)ws[NT + ci];
  float pv   = (float)ws[ci];
  float tmv  = (float)ws[2 * NT];
  float tmtv = (float)ws[2 * NT + 1];
  if (nc >= NT) { tpv = 0.f; pv = 0.f; tmv = 0.f; tmtv = 0.f; }

  // B 4x16 layout (f32): VGPR0 -> K = hi*2, VGPR1 -> K = hi*2+1.
  // Rows: K0=tp, K1=p, K2=sum_m, K3=sum_mt.
  v2f b;
  b.x = hi ? tmv  : tpv;
  b.y = hi ? tmtv : pv;

  // A 16x4 layout (f32): lane M = lane&15; VGPR0 -> K = hi*2, VGPR1 -> K=hi*2+1.
  const int m = nc;  // row index M
  const float c0 = (m < 4) ? ((m & 1) ? -1.f : 1.f) : 0.f;          // W[m][0]
  const float c1 = (m == 1) ? 1.f : ((m == 2) ? -1.f : 0.f);        // W[m][1]
  const float c2 = (m == 2) ? 1.f : 0.f;                            // W[m][2]
  const float c3 = (m == 2) ? -1.f : ((m == 3) ? 1.f : 0.f);        // W[m][3]
  v2f a;
  a.x = hi ? c2 : c0;
  a.y = hi ? c3 : c1;

  v8f c = {};
  // 8 args: (neg_a, A, neg_b, B, c_mod, C, reuse_a, reuse_b)
  v8f d = __builtin_amdgcn_wmma_f32_16x16x4_f32(
      false, a, false, b, (short)0, c, false, false);

  // D 16x16 layout: VGPR v, lanes 0-15 -> row M=v, col N=lane.
  // Rows 0..3 x cols 0..9 are the answer.
  if (hi == 0 && nc < NT) {
#pragma unroll
    for (int r = 0; r < 4; ++r) out[r * NT + nc] = d[r];
  }
#else
  // Scalar fallback (should not trigger on gfx1250 toolchains).
  const int nc = threadIdx.x;
  if (nc < NT) {
    const float tp  = (float)ws[NT + nc];
    const float p   = (float)ws[nc];
    const float tm  = (float)ws[2 * NT];
    const float tmt = (float)ws[2 * NT + 1];
    const float fp = p - tp;
    const float fn = tmt - tp;
    const float tn = tm - p - fn;
    out[0 * NT + nc] = tp;
    out[1 * NT + nc] = fp;
    out[2 * NT + nc] = tn;
    out[3 * NT + nc] = fn;
  }
#endif
}

extern "C" void kernel_launch(void* const* d_in, const int* in_sizes, int n_in,
                              void* d_out, int out_size, void* d_ws, size_t ws_size,
                              hipStream_t stream) {
  const float* pred = (const float*)d_in[0];
  const int*   mask = (const int*)d_in[1];
  const int*   targ = (const int*)d_in[2];
  float*       out  = (float*)d_out;
  unsigned*    ws   = (unsigned*)d_ws;
  const long long n = (long long)in_sizes[0];

  // Zero the 22 u32 global counters (graph-capturable memset node).
  hipMemsetAsync(d_ws, 0, (2 * NT + 2) * sizeof(unsigned), stream);

  const int threads = 256;           // 8 wave32 per block
  int blocks = 2048;                 // grid-stride; ~16 float4 per thread
  const long long nv = n >> 2;
  if ((long long)blocks * threads > nv && nv > 0) {
    blocks = (int)((nv + threads - 1) / threads);
  }
  if (blocks < 1) blocks = 1;

  precrec_count_kernel<<<blocks, threads, 0, stream>>>(pred, mask, targ, ws, n);
  precrec_finalize_kernel<<<1, 32, 0, stream>>>(ws, out);
}